// StreamingMultiHeadAttention_17987323036533
// MI455X (gfx1250) — compile-verified
//
#include <hip/hip_runtime.h>
#include <math.h>

#define B_      32
#define H_      32
#define DM_     4096
#define DK_     128
#define SINK_   4
#define WINDOW_ 1020
#define MAXC_   1024

typedef float v2f __attribute__((ext_vector_type(2)));
typedef float v8f __attribute__((ext_vector_type(8)));
typedef int   v4i __attribute__((vector_size(16)));

#if defined(__has_builtin)
#if __has_builtin(__builtin_amdgcn_global_load_async_to_lds_b128) && \
    __has_builtin(__builtin_amdgcn_s_wait_asynccnt)
#define HAVE_ASYNC_LDS 1
#endif
#endif
#ifndef HAVE_ASYNC_LDS
#define HAVE_ASYNC_LDS 0
#endif

// ---------------------------------------------------------------------------
// One wave computes a 16x16 tile of OUT = X (32 x DM) * W^T (DM x DM).
// f32 WMMA 16x16x4:
//   A (16x4):  lanes 0-15 -> rows m0+l, K = k0,k0+1 ; lanes 16-31 -> K = k0+2,k0+3
//   B (4x16):  lanes 0-15 -> cols n0+l, K = k0,k0+1 ; lanes 16-31 -> K = k0+2,k0+3
//   (B[k][n] = W[n][k], so W rows are streamed contiguously, read exactly once.)
// ---------------------------------------------------------------------------
__device__ __forceinline__ void gemm_tile_f32(const float* __restrict__ X,
                                              const float* __restrict__ W,
                                              float* __restrict__ OUT,
                                              int m0, int n0, int lane)
{
    const int half = lane >> 4;   // 0: K lo pair, 1: K hi pair
    const int l16  = lane & 15;
    const float* pa = X + (m0 + l16) * DM_ + 2 * half;
    const float* pb = W + (n0 + l16) * DM_ + 2 * half;

    v8f acc = {};
#pragma unroll 4
    for (int k = 0; k < DM_; k += 4) {
        v2f a = *(const v2f*)(pa + k);
        v2f b = *(const v2f*)(pb + k);
        // (neg_a, A, neg_b, B, c_mod, C, reuse_a, reuse_b)
        acc = __builtin_amdgcn_wmma_f32_16x16x4_f32(false, a, false, b,
                                                    (short)0, acc, false, false);
    }
    // D layout: VGPR j -> (lanes 0-15: M=j, N=l16) (lanes 16-31: M=8+j, N=l16)
#pragma unroll
    for (int j = 0; j < 8; ++j) {
        int row = m0 + j + half * 8;
        OUT[row * DM_ + n0 + l16] = acc[j];
    }
}

// ---------------------------------------------------------------------------
// Kernel 1: Q/K/V projections.  1536 waves = 3 mats x 2 M-tiles x 256 N-tiles
// ---------------------------------------------------------------------------
__global__ __launch_bounds__(256)
void qkv_gemm_kernel(const float* __restrict__ x,
                     const float* __restrict__ wq,
                     const float* __restrict__ wk,
                     const float* __restrict__ wv,
                     float* __restrict__ qb,
                     float* __restrict__ kb,
                     float* __restrict__ vb)
{
    const int wave = (int)((blockIdx.x * blockDim.x + threadIdx.x) >> 5);
    const int lane = (int)(threadIdx.x & 31);
    const int mat  = wave >> 9;          // 0..2
    const int rem  = wave & 511;
    const int m0   = (rem >> 8) << 4;    // 0 or 16
    const int n0   = (rem & 255) << 4;   // 0..4080
    const float* W = (mat == 0) ? wq : (mat == 1) ? wk : wv;
    float* OUT     = (mat == 0) ? qb : (mat == 1) ? kb : vb;
    gemm_tile_f32(x, W, OUT, m0, n0, lane);
}

// ---------------------------------------------------------------------------
// Kernel 3: output projection  OUT = O (32 x DM) * W_O^T.  512 waves.
// ---------------------------------------------------------------------------
__global__ __launch_bounds__(256)
void out_gemm_kernel(const float* __restrict__ ob,
                     const float* __restrict__ wo,
                     float* __restrict__ out)
{
    const int wave = (int)((blockIdx.x * blockDim.x + threadIdx.x) >> 5);
    const int lane = (int)(threadIdx.x & 31);
    const int m0   = (wave >> 8) << 4;
    const int n0   = (wave & 255) << 4;
    gemm_tile_f32(ob, wo, out, m0, n0, lane);
}

// ---------------------------------------------------------------------------
// Active position p -> cache row (or -1 for the freshly projected K/V).
// ---------------------------------------------------------------------------
__device__ __forceinline__ int map_pos(int p, int L, int seq, int insert, int tail)
{
    if (p == L - 1) return -1;                 // the newly inserted row
    if (seq < MAXC_) return p;                 // no wrap yet
    if (p < SINK_) return p;                   // sink tokens
    if (p < SINK_ + tail) return insert + 1 + (p - SINK_);
    return p - tail;                           // SINK_..insert-1
}

#if HAVE_ASYNC_LDS
typedef __attribute__((address_space(1))) v4i* gas_v4i;
typedef __attribute__((address_space(3))) v4i* las_v4i;

// Copy one 512B K/V row (128 floats) global -> LDS with one async b128 per lane.
__device__ __forceinline__ void async_copy_row(const float* g, float* lds_dst, int lane)
{
    float* gnc = const_cast<float*>(g + 4 * lane);
    __builtin_amdgcn_global_load_async_to_lds_b128(
        (gas_v4i)gnc,
        (las_v4i)(lds_dst + 4 * lane),
        0, 0);
}
#endif

// ---------------------------------------------------------------------------
// Kernel 2: streaming attention for one (b,h) per workgroup (8 waves).
// Pass 1: rope(K)·rope(Q) scores -> LDS ; softmax ; Pass 2: weighted V sum.
// K/V rows are streamed via double-buffered GLOBAL_LOAD_ASYNC_TO_LDS_B128
// (ASYNCcnt pipeline) when the toolchain exposes the builtin.
// ---------------------------------------------------------------------------
__global__ __launch_bounds__(256)
void attn_kernel(const float* __restrict__ kcache,
                 const float* __restrict__ vcache,
                 const float* __restrict__ qb,
                 const float* __restrict__ kb,
                 const float* __restrict__ vb,
                 const int*   __restrict__ seqp,
                 float* __restrict__ ob)
{
    __shared__ float s_sc[MAXC_];
    __shared__ float s_red[256];
    __shared__ float s_acc[DK_];
#if HAVE_ASYNC_LDS
    __shared__ float s_stage[8][2][DK_];   // per-wave double buffer (8KB)
#endif

    const int tid  = (int)threadIdx.x;
    const int lane = tid & 31;
    const int w    = tid >> 5;            // wave id 0..7
    const int bh   = (int)blockIdx.x;
    const int b    = bh >> 5;             // / H_
    const int h    = bh & 31;

    const int seq = seqp[0];
    int L, insert;
    if (seq < MAXC_) { L = seq + 1; insert = seq; }
    else             { L = MAXC_;  insert = SINK_ + (seq - SINK_) % WINDOW_; }
    const int tail = MAXC_ - 1 - insert;

    // Per-lane RoPE angles: lane owns elements (2l, 2l+1) and (64+2l, 64+2l+1)
    const float tha = __powf(10000.0f, -(float)(2 * lane)     * (1.0f / 64.0f));
    const float thb = __powf(10000.0f, -(float)(2 * lane + 1) * (1.0f / 64.0f));

    // Rotate Q at position L-1
    const float* qrow = qb + b * DM_ + h * DK_;
    float2 ql = *(const float2*)(qrow + 2 * lane);
    float2 qh = *(const float2*)(qrow + 64 + 2 * lane);
    {
        const float mq = (float)(L - 1);
        const float ca = __cosf(mq * tha), sa = __sinf(mq * tha);
        const float cb = __cosf(mq * thb), sb = __sinf(mq * thb);
        float r0 = ql.x * ca - qh.x * sa;
        float r1 = ql.y * cb - qh.y * sb;
        float r2 = qh.x * ca + ql.x * sa;
        float r3 = qh.y * cb + ql.y * sb;
        ql.x = r0; ql.y = r1; qh.x = r2; qh.y = r3;
    }

    const float* knew  = kb + b * DM_ + h * DK_;
    const float* vnew  = vb + b * DM_ + h * DK_;
    const float* kbase = kcache + (size_t)(b * H_ + h) * MAXC_ * DK_;
    const float* vbase = vcache + (size_t)(b * H_ + h) * MAXC_ * DK_;
    const float  scale = 0.0883883476483184f;   // 1/sqrt(128)

    // ---- Pass 1: scores ----------------------------------------------------
#if HAVE_ASYNC_LDS
    {
        int ibuf = 0;
        if (w < L) {
            const int s0 = map_pos(w, L, seq, insert, tail);
            const float* g0 = (s0 < 0) ? knew : (kbase + (size_t)s0 * DK_);
            async_copy_row(g0, &s_stage[w][0][0], lane);
        }
        for (int p = w; p < L; p += 8) {
            const int pn = p + 8;
            if (pn < L) {
                const int sn = map_pos(pn, L, seq, insert, tail);
                const float* gn = (sn < 0) ? knew : (kbase + (size_t)sn * DK_);
                async_copy_row(gn, &s_stage[w][ibuf ^ 1][0], lane);
                __builtin_amdgcn_s_wait_asynccnt(1);   // older copy complete
            } else {
                __builtin_amdgcn_s_wait_asynccnt(0);
            }
            const float* krow = &s_stage[w][ibuf][0];
            float2 kl = *(const float2*)(krow + 2 * lane);
            float2 kh = *(const float2*)(krow + 64 + 2 * lane);
            const float mp  = (float)p;
            const float cka = __cosf(mp * tha), ska = __sinf(mp * tha);
            const float ckb = __cosf(mp * thb), skb = __sinf(mp * thb);
            const float r0 = kl.x * cka - kh.x * ska;
            const float r1 = kl.y * ckb - kh.y * skb;
            const float r2 = kh.x * cka + kl.x * ska;
            const float r3 = kh.y * ckb + kl.y * skb;
            float part = ql.x * r0 + ql.y * r1 + qh.x * r2 + qh.y * r3;
#pragma unroll
            for (int off = 16; off > 0; off >>= 1)
                part += __shfl_down(part, off, 32);
            if (lane == 0) s_sc[p] = part * scale;
            ibuf ^= 1;
        }
        __builtin_amdgcn_s_wait_asynccnt(0);
    }
#else
    for (int p = w; p < L; p += 8) {
        const int src = map_pos(p, L, seq, insert, tail);
        const float* krow = (src < 0) ? knew : (kbase + (size_t)src * DK_);
        const int pn = p + 8;
        if (pn < L) {
            const int sn = map_pos(pn, L, seq, insert, tail);
            const float* krn = (sn < 0) ? knew : (kbase + (size_t)sn * DK_);
            __builtin_prefetch(krn + 2 * lane, 0, 0);
        }
        float2 kl = *(const float2*)(krow + 2 * lane);
        float2 kh = *(const float2*)(krow + 64 + 2 * lane);
        const float mp  = (float)p;
        const float cka = __cosf(mp * tha), ska = __sinf(mp * tha);
        const float ckb = __cosf(mp * thb), skb = __sinf(mp * thb);
        const float r0 = kl.x * cka - kh.x * ska;
        const float r1 = kl.y * ckb - kh.y * skb;
        const float r2 = kh.x * cka + kl.x * ska;
        const float r3 = kh.y * ckb + kl.y * skb;
        float part = ql.x * r0 + ql.y * r1 + qh.x * r2 + qh.y * r3;
#pragma unroll
        for (int off = 16; off > 0; off >>= 1)
            part += __shfl_down(part, off, 32);
        if (lane == 0) s_sc[p] = part * scale;
    }
#endif
    __syncthreads();

    // ---- softmax (block reduction) ----------------------------------------
    float lm = -3.4e38f;
    for (int p = tid; p < L; p += 256) lm = fmaxf(lm, s_sc[p]);
    s_red[tid] = lm;
    __syncthreads();
    for (int st = 128; st > 0; st >>= 1) {
        if (tid < st) s_red[tid] = fmaxf(s_red[tid], s_red[tid + st]);
        __syncthreads();
    }
    const float gmax = s_red[0];
    __syncthreads();

    float ls = 0.0f;
    for (int p = tid; p < L; p += 256) {
        const float e = __expf(s_sc[p] - gmax);
        s_sc[p] = e;
        ls += e;
    }
    s_red[tid] = ls;
    __syncthreads();
    for (int st = 128; st > 0; st >>= 1) {
        if (tid < st) s_red[tid] += s_red[tid + st];
        __syncthreads();
    }
    const float inv = 1.0f / s_red[0];
    if (tid < DK_) s_acc[tid] = 0.0f;
    __syncthreads();

    // ---- Pass 2: weighted V accumulation ----------------------------------
    float a0 = 0.f, a1 = 0.f, a2 = 0.f, a3 = 0.f;
#if HAVE_ASYNC_LDS
    {
        int ibuf = 0;
        if (w < L) {
            const int s0 = map_pos(w, L, seq, insert, tail);
            const float* g0 = (s0 < 0) ? vnew : (vbase + (size_t)s0 * DK_);
            async_copy_row(g0, &s_stage[w][0][0], lane);
        }
        for (int p = w; p < L; p += 8) {
            const int pn = p + 8;
            if (pn < L) {
                const int sn = map_pos(pn, L, seq, insert, tail);
                const float* gn = (sn < 0) ? vnew : (vbase + (size_t)sn * DK_);
                async_copy_row(gn, &s_stage[w][ibuf ^ 1][0], lane);
                __builtin_amdgcn_s_wait_asynccnt(1);
            } else {
                __builtin_amdgcn_s_wait_asynccnt(0);
            }
            const float4 v = *(const float4*)(&s_stage[w][ibuf][0] + 4 * lane);
            const float wt = s_sc[p];
            a0 += wt * v.x; a1 += wt * v.y; a2 += wt * v.z; a3 += wt * v.w;
            ibuf ^= 1;
        }
        __builtin_amdgcn_s_wait_asynccnt(0);
    }
#else
    for (int p = w; p < L; p += 8) {
        const int src = map_pos(p, L, seq, insert, tail);
        const float* vrow = (src < 0) ? vnew : (vbase + (size_t)src * DK_);
        const float4 v = *(const float4*)(vrow + 4 * lane);
        const float wt = s_sc[p];
        a0 += wt * v.x; a1 += wt * v.y; a2 += wt * v.z; a3 += wt * v.w;
    }
#endif
    atomicAdd(&s_acc[4 * lane + 0], a0);
    atomicAdd(&s_acc[4 * lane + 1], a1);
    atomicAdd(&s_acc[4 * lane + 2], a2);
    atomicAdd(&s_acc[4 * lane + 3], a3);
    __syncthreads();

    if (tid < DK_) ob[b * DM_ + h * DK_ + tid] = s_acc[tid] * inv;
}

// ---------------------------------------------------------------------------
extern "C" void kernel_launch(void* const* d_in, const int* in_sizes, int n_in,
                              void* d_out, int out_size, void* d_ws, size_t ws_size,
                              hipStream_t stream)
{
    const float* x  = (const float*)d_in[0];
    const float* kc = (const float*)d_in[1];
    const float* vc = (const float*)d_in[2];
    const float* wq = (const float*)d_in[3];
    const float* wk = (const float*)d_in[4];
    const float* wv = (const float*)d_in[5];
    const float* wo = (const float*)d_in[6];
    const int*   sq = (const int*)d_in[7];

    float* ws = (float*)d_ws;
    float* qb = ws;                        // 32*4096
    float* kb = ws + 1 * B_ * DM_;         // K_new rows
    float* vb = ws + 2 * B_ * DM_;         // V_new rows
    float* ob = ws + 3 * B_ * DM_;         // attention output

    // 1536 waves (3 mats x 2 M-tiles x 256 N-tiles) -> 192 blocks of 8 waves
    qkv_gemm_kernel<<<192, 256, 0, stream>>>(x, wq, wk, wv, qb, kb, vb);

    // one workgroup per (b,h)
    attn_kernel<<<B_ * H_, 256, 0, stream>>>(kc, vc, qb, kb, vb, sq, ob);

    // 512 waves -> 64 blocks
    out_gemm_kernel<<<64, 256, 0, stream>>>(ob, wo, (float*)d_out);
}